// Conv_8443905704574
// MI455X (gfx1250) — compile-verified
//
#include <hip/hip_runtime.h>

// CDNA5 / gfx1250: wave32, WMMA f32 16x16x4, async-b128 global->LDS staging,
// non-temporal b128 stores.
typedef __attribute__((ext_vector_type(2))) float v2f;
typedef __attribute__((ext_vector_type(4))) float v4f;
typedef __attribute__((ext_vector_type(8))) float v8f;

namespace cfg {
constexpr int BATCH = 16;
constexpr int CHANS = 3;
constexpr int IMG_H = 1024;
constexpr int IMG_W = 1024;
constexpr int KS    = 7;
constexpr int PAD   = 3;     // KS/2

constexpr int TM = 64;                   // output rows per workgroup (4 tiles/wave)
constexpr int TN = 128;                  // output cols per workgroup (8 waves x 16)
constexpr int NT_ROW = 4;                // row-tiles per wave
constexpr int HALO_ROWS  = TM + KS - 1;     // 70
constexpr int STG_COLS   = 136;             // staged cols: x0-4 .. x0+131 (f4-aligned)
constexpr int LDS_STRIDE = 140;             // padded input row stride (floats), 16B-mult
constexpr int NKC  = 6;                     // K chunks of 4 per dy (K = 24)
constexpr int TSTR = 132;                   // transpose scratch stride (floats)
}

__global__ __launch_bounds__(256)
void conv7x7_wmma_f32(const float* __restrict__ img,
                      const float* __restrict__ ker,
                      float* __restrict__ out)
{
  using namespace cfg;

  // Input tile + halo; LDS col 0 corresponds to global x = x0 - 4.
  // Later reused as the store-transpose scratch (64*132 = 8448 <= 70*140 = 9800).
  __shared__ __align__(16) float sIn[HALO_ROWS * LDS_STRIDE];   // 39200 B
  // Banded weight matrices in WMMA B-fragment order. With the A-slab origin
  // at x0w-4 (k = 0..23, LDS col = xw + k):
  //   out[x0w+n] needs w[dy][dx] * in[x0w + n + dx - 3]  =>  k = n + dx + 1,
  //   so B[k][n] = w[dy][k-n-1] if 0 <= k-n-1 < 7, else 0 (k=0,23 are zero).
  // slot ((dy*NKC + c)*32 + lane)*2 + v holds B[k][n], k = 4c + 2h + v,
  // n = lane&15, h = lane>>4.
  __shared__ float sB[KS * NKC * 64];                           // 10752 B

  const int tid  = threadIdx.x;
  const int lane = tid & 31;
  const int wav  = tid >> 5;

  const int x0 = blockIdx.x * TN;
  const int y0 = blockIdx.y * TM;
  const int b  = blockIdx.z;

  const float* src = img + (size_t)(b * CHANS + (CHANS - 1)) * IMG_H * IMG_W;

  // Uniform per-workgroup branch; interior needs rows y0-3..y0+66 and
  // cols x0-4..x0+131 fully inside the image.
  const bool interior = (y0 >= PAD) && (y0 + TM - 1 + PAD < IMG_H) &&
                        (x0 >= 4)   && (x0 + TN + PAD < IMG_W);

  if (interior) {
    // Async global -> LDS staging, 16B per lane per op (ASYNCcnt-tracked).
    // 70 rows x 34 float4-chunks = 2380 chunks. All compute reads fall inside
    // the staged 136 columns, so no zero-fill is needed on this path.
    const unsigned ldsbase = (unsigned)(size_t)(void*)sIn;
    const float* gbase = src + (size_t)(y0 - PAD) * IMG_W + (x0 - 4);
    constexpr int CHUNKS = STG_COLS / 4;           // 34
    for (int idx = tid; idx < HALO_ROWS * CHUNKS; idx += 256) {
      const int r  = idx / CHUNKS;
      const int cc = idx - r * CHUNKS;
      const unsigned ldsaddr = ldsbase + (unsigned)(r * LDS_STRIDE + cc * 4) * 4u;
      const unsigned long long gaddr =
          (unsigned long long)(const void*)(gbase + (size_t)r * IMG_W + cc * 4);
      asm volatile("global_load_async_to_lds_b128 %0, %1, off"
                   :: "v"(ldsaddr), "v"(gaddr)
                   : "memory");
    }
    asm volatile("s_wait_asynccnt 0x0" ::: "memory");
  } else {
    // Border tiles: bounds-checked scalar staging with zero padding.
    for (int idx = tid; idx < HALO_ROWS * LDS_STRIDE; idx += 256) {
      const int r   = idx / LDS_STRIDE;
      const int col = idx - r * LDS_STRIDE;
      const int gy  = y0 - PAD + r;
      const int gx  = x0 - 4 + col;
      float v = 0.0f;
      if (col < STG_COLS && gy >= 0 && gy < IMG_H && gx >= 0 && gx < IMG_W)
        v = src[(size_t)gy * IMG_W + gx];
      sIn[idx] = v;
    }
  }

  // Banded-B fragments (shared by all four row-tiles of every wave).
  for (int s = tid; s < KS * NKC * 32; s += 256) {
    const int dy  = s / (NKC * 32);
    const int rem = s - dy * (NKC * 32);
    const int c   = rem >> 5;
    const int l   = rem & 31;
    const int n   = l & 15;
    const int h   = l >> 4;
#pragma unroll
    for (int v = 0; v < 2; ++v) {
      const int k = 4 * c + 2 * h + v;
      const int d = k - n - 1;               // band shifted by the -4 A origin
      sB[(s << 1) | v] = (d >= 0 && d < KS) ? ker[dy * KS + d] : 0.0f;
    }
  }

  __syncthreads();

  // ---- Compute: 168 x v_wmma_f32_16x16x4_f32 per wave (4 row-tiles / B) ----
  const int n  = lane & 15;    // N column (B/C/D) == A row M
  const int h  = lane >> 4;    // lane half: K select (A/B), M+8 (C/D)
  const int xw = wav * 16;     // wave's column offset inside the WG tile

  v8f acc[NT_ROW];
#pragma unroll
  for (int t = 0; t < NT_ROW; ++t)
    acc[t] = (v8f){0.f, 0.f, 0.f, 0.f, 0.f, 0.f, 0.f, 0.f};

#pragma unroll
  for (int dy = 0; dy < KS; ++dy) {
    const float* rp[NT_ROW];
#pragma unroll
    for (int t = 0; t < NT_ROW; ++t)
      rp[t] = &sIn[(n + 16 * t + dy) * LDS_STRIDE];
#pragma unroll
    for (int c = 0; c < NKC; ++c) {
      const float* bp = &sB[((((dy * NKC + c) << 5) | lane) << 1)];
      v2f bb;  bb.x = bp[0];  bb.y = bp[1];
      const int colb = xw + 4 * c + 2 * h;   // even => 8B-aligned pair loads
#pragma unroll
      for (int t = 0; t < NT_ROW; ++t) {
        v2f a;  a.x = rp[t][colb];  a.y = rp[t][colb + 1];
        acc[t] = __builtin_amdgcn_wmma_f32_16x16x4_f32(
            false, a, false, bb, (short)0, acc[t], false, false);
      }
    }
  }

  // ---- Transpose C fragments through LDS, then wide non-temporal stores ----
  __syncthreads();             // input staging no longer needed; reuse sIn
  float* T = sIn;              // 64 rows x 128 cols, stride TSTR=132

#pragma unroll
  for (int t = 0; t < NT_ROW; ++t)
#pragma unroll
    for (int r = 0; r < 8; ++r)
      T[(16 * t + r + 8 * h) * TSTR + xw + n] = acc[t][r];

  __syncthreads();

  // 64 rows x 32 chunks of 4 floats = 2048 chunks; 8 per thread; one wave
  // covers one full 512B row segment per instruction. Broadcast to 3 channels
  // with TH=NT (pure streaming, no reuse).
  float* o0 = out + (size_t)(b * CHANS + 0) * IMG_H * IMG_W;
  float* o1 = out + (size_t)(b * CHANS + 1) * IMG_H * IMG_W;
  float* o2 = out + (size_t)(b * CHANS + 2) * IMG_H * IMG_W;
#pragma unroll
  for (int i = tid; i < TM * 32; i += 256) {
    const int row = i >> 5;
    const int cb  = i & 31;
    const v4f val = *(const v4f*)&T[row * TSTR + cb * 4];
    const size_t sp = (size_t)(y0 + row) * IMG_W + (x0 + cb * 4);
    __builtin_nontemporal_store(val, (v4f*)&o0[sp]);
    __builtin_nontemporal_store(val, (v4f*)&o1[sp]);
    __builtin_nontemporal_store(val, (v4f*)&o2[sp]);
  }
}

extern "C" void kernel_launch(void* const* d_in, const int* in_sizes, int n_in,
                              void* d_out, int out_size, void* d_ws, size_t ws_size,
                              hipStream_t stream) {
  using namespace cfg;
  const float* img = (const float*)d_in[0];   // (16,3,1024,1024) f32
  const float* ker = (const float*)d_in[1];   // (7,7) f32
  float* out = (float*)d_out;                 // (16,3,1024,1024) f32

  dim3 grid(IMG_W / TN, IMG_H / TM, BATCH);   // (8, 16, 16)
  dim3 block(256);                            // 8 wave32 waves
  conv7x7_wmma_f32<<<grid, block, 0, stream>>>(img, ker, out);
}